// MoE_v1_37074157699448
// MI455X (gfx1250) — compile-verified
//
#include <hip/hip_runtime.h>

typedef float v2f __attribute__((ext_vector_type(2)));
typedef float v8f __attribute__((ext_vector_type(8)));

#define KK 16
#define BB 8
#define CC 3
#define WW 384
#define HH 384

__device__ __forceinline__ float softplus_f(float x) {
    // numerically stable: max(x,0) + log1p(exp(-|x|))
    return fmaxf(x, 0.0f) + log1pf(__expf(-fabsf(x)));
}

// ---------------------------------------------------------------------------
// Prep: per (b,c,k) derive  r0, r1, m0, m1, wsm  into workspace, where
//   r0 = -0.5*log2(e)*(S11-S01)/det,  r1 = -0.5*log2(e)*(S00-S01)/det
// (row-sums of Sigma^-1, pre-scaled so the main kernel's exponents are in the
// log2 domain -> a single raw v_exp_f32 per softmax element).
// One wave per (b,c); lanes 0-15 handle k, softmax over w via shfl_xor.
// ---------------------------------------------------------------------------
__global__ void gsplat_prep(const float* __restrict__ params, float* __restrict__ ws) {
    int bc = blockIdx.x;            // 0..23
    int lane = threadIdx.x;         // 0..31
    int k = lane & 15;
    const float* p = params + bc * (7 * KK);
    float m0  = p[k];
    float m1  = p[KK + k];
    float w   = p[2 * KK + k];
    float sr0 = p[3 * KK + 2 * k];
    float sr1 = p[3 * KK + 2 * k + 1];
    float th  = p[5 * KK + k];
    // theta wrap to [-pi,pi) leaves cos/sin unchanged -> skip it
    float s0 = softplus_f(sr0) + 1e-8f + 1e-4f;   // scale + EPS_S
    float s1 = softplus_f(sr1) + 1e-8f + 1e-4f;
    float ct = cosf(th), st = sinf(th);
    float c2 = ct * ct, s2 = st * st;
    float q0 = s0 * s0, q1 = s1 * s1;
    float S00 = c2 * q0 + s2 * q1;
    float S11 = s2 * q0 + c2 * q1;
    float S01 = ct * st * (q0 - q1);
    float det = S00 * S11 - S01 * S01;
    const float nhl2e = -0.5f * 1.44269504088896340736f;  // -0.5*log2(e)
    float r0 = nhl2e * (S11 - S01) / det;
    float r1 = nhl2e * (S00 - S01) / det;
    // softmax of w over the 16 components (groups of 16 lanes)
    float wm = w;
    for (int m = 8; m >= 1; m >>= 1) wm = fmaxf(wm, __shfl_xor(wm, m, 16));
    float e = __expf(w - wm);
    float esum = e;
    for (int m = 8; m >= 1; m >>= 1) esum += __shfl_xor(esum, m, 16);
    float wsm = e / esum;
    if (lane < 16) {
        float* q = ws + (size_t)(bc * KK + k) * 8;
        q[0] = m0; q[1] = m1; q[2] = r0; q[3] = r1; q[4] = wsm;
    }
}

// ---------------------------------------------------------------------------
// Main: one wave per 16x16 pixel tile of one (b,c) image.
// Per Gaussian k:  log2-exponent tile = A(16x4) x B(4x16) via
// v_wmma_f32_16x16x4_f32:
//   A (lanes 0-15: K=0,1 | lanes 16-31: K=2,3):
//       [ r0*d0^2 , (r0+r1)*d0 | r1 , 0 ]
//   B: [ 1 , t | t^2 , 0 ]        with d0 = u - m0, t = v - m1
// Then elementwise softmax (raw v_exp_f32) over the 16 accumulators + mix.
// ---------------------------------------------------------------------------
__global__ void __launch_bounds__(256) gsplat_main(const float* __restrict__ ws,
                                                   float* __restrict__ out) {
    int lane = threadIdx.x & 31;
    int wave = threadIdx.x >> 5;
    int bc = blockIdx.z;
    int x0 = blockIdx.x * 16;
    int y0 = (blockIdx.y * 8 + wave) * 16;
    int lm = lane & 15;
    bool lo = lane < 16;
    const float inv = 1.0f / (float)(WW - 1);   // linspace(0,1,384)
    float u = (float)(x0 + lm) * inv;           // x coordinate (row of A / M)
    float v = (float)(y0 + lm) * inv;           // y coordinate (col of B / N)

    const float* pbase = ws + (size_t)bc * KK * 8;
    v8f acc[KK];
    v8f zero = {};
#pragma unroll
    for (int k = 0; k < KK; ++k) {
        const float* p = pbase + k * 8;
        float m0 = p[0], m1 = p[1], r0 = p[2], r1 = p[3];
        float d0 = u - m0;
        float t  = v - m1;
        float aA = r0 * d0 * d0;
        float aB = (r0 + r1) * d0;
        v2f A, Bv;
        A.x  = lo ? aA   : r1;       // K=0 | K=2
        A.y  = lo ? aB   : 0.0f;     // K=1 | K=3
        Bv.x = lo ? 1.0f : t * t;    // K=0 | K=2
        Bv.y = lo ? t    : 0.0f;     // K=1 | K=3
        acc[k] = __builtin_amdgcn_wmma_f32_16x16x4_f32(
            false, A, false, Bv, (short)0, zero, false, false);
    }

    // per-pixel max over k (elementwise across the 16 accumulators)
    float mx[8];
#pragma unroll
    for (int j = 0; j < 8; ++j) mx[j] = acc[0][j];
#pragma unroll
    for (int k = 1; k < KK; ++k)
#pragma unroll
        for (int j = 0; j < 8; ++j) mx[j] = fmaxf(mx[j], acc[k][j]);

    float num[8], den[8];
#pragma unroll
    for (int j = 0; j < 8; ++j) { num[j] = 0.0f; den[j] = 0.0f; }
#pragma unroll
    for (int k = 0; k < KK; ++k) {
        float wk = pbase[k * 8 + 4];
#pragma unroll
        for (int j = 0; j < 8; ++j) {
            // raw v_exp_f32; args are <= 0 and FTZ of tiny terms is harmless
            float e = __builtin_amdgcn_exp2f(acc[k][j] - mx[j]);
            den[j] += e;
            num[j] = fmaf(e, wk, num[j]);
        }
    }

    // D layout: VGPR j -> x-row (j + 8*(lane>=16)), lane&15 -> y-column
    int y  = y0 + lm;
    int xh = x0 + ((lane >> 4) << 3);
    float* o = out + (size_t)bc * WW * HH + y;
#pragma unroll
    for (int j = 0; j < 8; ++j) {
        float val = num[j] / (den[j] + 1e-8f);
        val = fminf(fmaxf(val, 0.0f), 1.0f);
        o[(size_t)(xh + j) * HH] = val;
    }
}

extern "C" void kernel_launch(void* const* d_in, const int* in_sizes, int n_in,
                              void* d_out, int out_size, void* d_ws, size_t ws_size,
                              hipStream_t stream) {
    const float* params = (const float*)d_in[0];  // (8,3,112) f32
    float* out = (float*)d_out;                   // (8,3,384,384) f32
    float* ws  = (float*)d_ws;                    // needs 24*16*8*4 = 12 KB

    gsplat_prep<<<dim3(BB * CC), dim3(32), 0, stream>>>(params, ws);
    gsplat_main<<<dim3(WW / 16, HH / (16 * 8), BB * CC), dim3(256), 0, stream>>>(ws, out);
}